// VGAEEncoder_1211180778233
// MI455X (gfx1250) — compile-verified
//
#include <hip/hip_runtime.h>

typedef __attribute__((ext_vector_type(2))) float v2f;
typedef __attribute__((ext_vector_type(8))) float v8f;

#define N_NODES 100000
#define IN_CH   256
#define HID_CH  128
#define OUT_CH  64

// ---------------- utility kernels ----------------

__global__ __launch_bounds__(256) void fill_kernel(float* __restrict__ p, float v, int n) {
  int i = blockIdx.x * blockDim.x + threadIdx.x;
  if (i < n) p[i] = v;
}

__global__ __launch_bounds__(256) void deg_kernel(const int* __restrict__ dst,
                                                  float* __restrict__ deg, int E) {
  int e = blockIdx.x * blockDim.x + threadIdx.x;
  if (e < E) atomicAdd(&deg[dst[e]], 1.0f);
}

__global__ __launch_bounds__(256) void dis_kernel(const float* __restrict__ deg,
                                                  float* __restrict__ dis, int N) {
  int i = blockIdx.x * blockDim.x + threadIdx.x;
  if (i < N) dis[i] = rsqrtf(deg[i]);   // deg >= 1 always (self loop)
}

__global__ __launch_bounds__(256) void relu_kernel(float* __restrict__ p, int n) {
  int i = blockIdx.x * blockDim.x + threadIdx.x;
  if (i < n) p[i] = fmaxf(p[i], 0.0f);
}

// ---------------- dense GEMM via V_WMMA_F32_16X16X4_F32 ----------------
// out[N x M] = X[N x K] @ W[K x M], row-major everywhere.
// blockDim.x = 32 * (M/16); wave w owns 16-col tile w; blockIdx.x owns 16-row tile.
// A-matrix layout (ISA 7.12.2, 32-bit 16x4): lane(0-15)=row m, K={0,1}; lanes 16-31: K={2,3}.
// B-matrix mirrors: lane half selects K pair, vgpr index = k%2, lane%16 = col n.
// C/D: vgpr v -> row (v + 8*(lane>=16)), col = lane%16.
__global__ void gemm_wmma_kernel(const float* __restrict__ X, const float* __restrict__ W,
                                 float* __restrict__ out, int K, int M) {
  const int wave = threadIdx.x >> 5;
  const int lane = threadIdx.x & 31;
  const int lm   = lane & 15;
  const int kh   = lane >> 4;                 // which K half {0,1} vs {2,3}
  const int row0 = blockIdx.x << 4;
  const int col0 = wave << 4;

  v8f acc = {};
  const float* Arow = X + (size_t)(row0 + lm) * K + 2 * kh;
  const float* Bcol = W + (size_t)(2 * kh) * M + col0 + lm;

  for (int k = 0; k < K; k += 4) {
    v2f a = *(const v2f*)(Arow + k);          // global_load_b64: A[m, k+2kh .. +1]
    v2f b;
    b.x = Bcol[(size_t)k * M];                // B[k+2kh,   n]
    b.y = Bcol[(size_t)k * M + M];            // B[k+2kh+1, n]
    acc = __builtin_amdgcn_wmma_f32_16x16x4_f32(
        /*neg_a=*/false, a, /*neg_b=*/false, b,
        /*c_mod=*/(short)0, acc, /*reuse_a=*/false, /*reuse_b=*/false);
  }

  float* o = out + (size_t)(row0 + 8 * kh) * M + col0 + lm;
#pragma unroll
  for (int v = 0; v < 8; ++v) o[(size_t)v * M] = acc[v];
}

// ---------------- GCN aggregation ----------------
// A[n,c] = bias[c] + dis[n]^2 * T[n,c]     (self-loop term + bias)
// blockDim = M, gridDim = N
__global__ void init_self_kernel(const float* __restrict__ T, const float* __restrict__ dis,
                                 const float* __restrict__ bias, float* __restrict__ A, int M) {
  int n = blockIdx.x;
  int c = threadIdx.x;
  float d = dis[n];
  A[(size_t)n * M + c] = bias[c] + d * d * T[(size_t)n * M + c];
}

// One wave per edge: coalesced read of T[src,:], coalesced atomic add into A[dst,:].
__global__ __launch_bounds__(256) void scatter_kernel(const int* __restrict__ src,
                                                      const int* __restrict__ dst,
                                                      const float* __restrict__ dis,
                                                      const float* __restrict__ T,
                                                      float* __restrict__ A, int E, int M) {
  int e    = blockIdx.x * 8 + (threadIdx.x >> 5);
  int lane = threadIdx.x & 31;
  if (e >= E) return;
  int s = src[e], d = dst[e];
  float nrm = dis[s] * dis[d];
  const float* Ts = T + (size_t)s * M;
  float*       Ad = A + (size_t)d * M;
  for (int c = lane; c < M; c += 32)
    atomicAdd(&Ad[c], nrm * Ts[c]);           // global_atomic_add_f32
}

// ---------------- driver ----------------

extern "C" void kernel_launch(void* const* d_in, const int* in_sizes, int n_in,
                              void* d_out, int out_size, void* d_ws, size_t ws_size,
                              hipStream_t stream) {
  const float* x    = (const float*)d_in[0];
  const int*   ei   = (const int*)  d_in[1];
  const float* W1   = (const float*)d_in[2];
  const float* b1   = (const float*)d_in[3];
  const float* W2   = (const float*)d_in[4];
  const float* b2   = (const float*)d_in[5];
  const float* Wmu  = (const float*)d_in[6];
  const float* bmu  = (const float*)d_in[7];
  const float* Wls  = (const float*)d_in[8];
  const float* bls  = (const float*)d_in[9];

  const int N = in_sizes[0] / IN_CH;       // 100000
  const int E = in_sizes[1] / 2;           // 1600000
  const int* srcI = ei;
  const int* dstI = ei + E;

  float* ws  = (float*)d_ws;
  float* deg = ws;                                   // N
  float* dis = ws + N;                               // N
  float* T   = ws + 2 * (size_t)N;                   // N x 128
  float* A   = ws + 2 * (size_t)N + (size_t)N * HID_CH; // N x 128

  float* mu = (float*)d_out;               // N x 64
  float* ls = (float*)d_out + (size_t)N * OUT_CH;

  const int rowTiles = N / 16;             // 6250 (N divisible by 16)
  const int eBlocks  = (E + 7) / 8;        // wave per edge, 8 waves/block

  // ---- normalization ----
  fill_kernel<<<(N + 255) / 256, 256, 0, stream>>>(deg, 1.0f, N);       // self loop
  deg_kernel<<<(E + 255) / 256, 256, 0, stream>>>(dstI, deg, E);
  dis_kernel<<<(N + 255) / 256, 256, 0, stream>>>(deg, dis, N);

  // ---- layer 1: h1 = relu(agg(x @ W1) + b1) ----
  gemm_wmma_kernel<<<rowTiles, 32 * (HID_CH / 16), 0, stream>>>(x, W1, T, IN_CH, HID_CH);
  init_self_kernel<<<N, HID_CH, 0, stream>>>(T, dis, b1, A, HID_CH);
  scatter_kernel<<<eBlocks, 256, 0, stream>>>(srcI, dstI, dis, T, A, E, HID_CH);
  relu_kernel<<<((size_t)N * HID_CH + 255) / 256, 256, 0, stream>>>(A, N * HID_CH);

  // ---- layer 2: h2 = relu(agg(h1 @ W2) + b2) ----
  gemm_wmma_kernel<<<rowTiles, 32 * (HID_CH / 16), 0, stream>>>(A, W2, T, HID_CH, HID_CH);
  init_self_kernel<<<N, HID_CH, 0, stream>>>(T, dis, b2, A, HID_CH);
  scatter_kernel<<<eBlocks, 256, 0, stream>>>(srcI, dstI, dis, T, A, E, HID_CH);
  relu_kernel<<<((size_t)N * HID_CH + 255) / 256, 256, 0, stream>>>(A, N * HID_CH);

  // ---- mu = agg(h2 @ W_mu) + b_mu ----
  gemm_wmma_kernel<<<rowTiles, 32 * (OUT_CH / 16), 0, stream>>>(A, Wmu, T, HID_CH, OUT_CH);
  init_self_kernel<<<N, OUT_CH, 0, stream>>>(T, dis, bmu, mu, OUT_CH);
  scatter_kernel<<<eBlocks, 256, 0, stream>>>(srcI, dstI, dis, T, mu, E, OUT_CH);

  // ---- logstd = agg(h2 @ W_logstd) + b_logstd ----
  gemm_wmma_kernel<<<rowTiles, 32 * (OUT_CH / 16), 0, stream>>>(A, Wls, T, HID_CH, OUT_CH);
  init_self_kernel<<<N, OUT_CH, 0, stream>>>(T, dis, bls, ls, OUT_CH);
  scatter_kernel<<<eBlocks, 256, 0, stream>>>(srcI, dstI, dis, T, ls, E, OUT_CH);
}